// Model_44014824849774
// MI455X (gfx1250) — compile-verified
//
#include <hip/hip_runtime.h>
#include <hip/hip_bf16.h>

typedef __attribute__((ext_vector_type(2))) float v2f;
typedef __attribute__((ext_vector_type(8))) float v8f;

#define NUN 100000
#define NVN 100000
#define HDIM 128
#define CCLS 16

// ---------------------------------------------------------------------------
// GEMM: out[M,128] = act(A)[M,K] @ W[K,128], row-major fp32.
// Block tile: 16*MT rows x 128 cols. 8 waves; wave w owns cols [16w, 16w+16)
// across all MT row-subtiles (B fragment reused MT times per k-step).
// V_WMMA_F32_16X16X4_F32, accumulate over K in steps of 4.
//   A-frag: lane l -> m = l&15, kp = (l>>4)*2 ; a = {A[m][k+kp], A[m][k+kp+1]}
//   B-frag: lane l -> n = l&15, kp = (l>>4)*2 ; b = {W[k+kp][n], W[k+kp+1][n]}
//   C/D:    lane l -> n = l&15 ; vgpr r -> row r + 8*(l>>4)
// RELU_IN=false: A tile staged with GLOBAL_LOAD_ASYNC_TO_LDS_B128 (ASYNCcnt),
//                bypassing VGPRs entirely.
// RELU_IN=true:  A tile staged through VALU with fused relu.
// ---------------------------------------------------------------------------
template <int K, int MT, bool RELU_IN>
__global__ __launch_bounds__(256) void gemm_wmma(const float* __restrict__ A,
                                                 const float* __restrict__ W,
                                                 float* __restrict__ out) {
  constexpr int PAD = 4;               // stride K+4 floats -> banks 4*m, conflict-free
  constexpr int ROWS = 16 * MT;
  __shared__ float atile[ROWS * (K + PAD)];

  const int m0 = blockIdx.x * ROWS;
  const int t = threadIdx.x;

  if constexpr (!RELU_IN) {
    // Async DMA stage: each lane copies 16B per issue from global address gaddr to LDS address lds_addr.
    // Chunks never straddle rows (K % 4 == 0); both sides 16B aligned
    // ((K+PAD)*4 = multiple of 16 bytes).
    for (int i = t; i < ROWS * K / 4; i += 256) {
      const int e = i * 4;
      const int r = e / K;
      const int c = e - r * K;
      const unsigned lds_addr =
          (unsigned)(uintptr_t)(atile + r * (K + PAD) + c);
      const unsigned long long gaddr =
          (unsigned long long)(uintptr_t)(A + (size_t)(m0 + r) * K + c);
      asm volatile("global_load_async_to_lds_b128 %0, %1, off"
                   :: "v"(lds_addr), "v"(gaddr) : "memory");
    }
    asm volatile("s_wait_asynccnt 0x0" ::: "memory");
  } else {
    // VALU stage with fused relu of the previous layer's accumulator.
    for (int i = t; i < ROWS * K; i += 256) {
      const int r = i / K;
      const int c = i - r * K;
      atile[r * (K + PAD) + c] =
          fmaxf(A[(size_t)(m0 + r) * K + c], 0.0f);
    }
  }
  __syncthreads();

  const int lane = t & 31;
  const int n0 = (t >> 5) * 16;
  const int m = lane & 15;
  const int n = lane & 15;
  const int kp = (lane >> 4) * 2;

  const float* wcol = W + n0 + n + (size_t)kp * HDIM;
  const float* arow = atile + m * (K + PAD) + kp;

  v8f acc[MT];
  #pragma unroll
  for (int s = 0; s < MT; ++s) acc[s] = (v8f){};

  #pragma unroll 2
  for (int k = 0; k < K; k += 4) {
    v2f b;
    b.x = wcol[(size_t)k * HDIM];
    b.y = wcol[(size_t)(k + 1) * HDIM];
    #pragma unroll
    for (int s = 0; s < MT; ++s) {
      const float* ap = arow + s * 16 * (K + PAD) + k;
      v2f a;
      a.x = ap[0];
      a.y = ap[1];
      acc[s] = __builtin_amdgcn_wmma_f32_16x16x4_f32(false, a, false, b,
                                                     (short)0, acc[s], false, false);
    }
  }

  #pragma unroll
  for (int s = 0; s < MT; ++s) {
    const int rowBase = m0 + s * 16 + (lane >> 4) * 8;
    #pragma unroll
    for (int r = 0; r < 8; ++r)
      out[(size_t)(rowBase + r) * HDIM + n0 + n] = acc[s][r];
  }
}

// ---------------------------------------------------------------------------
// COO SpMM scatter-add: acc[row[e], :] += val[e] * t[col[e], :]
// 32 lanes per edge, 4 columns per lane (float4 gather, 4 f32 atomics).
// ---------------------------------------------------------------------------
__global__ __launch_bounds__(256) void spmm_scatter(const int* __restrict__ row,
                                                    const int* __restrict__ col,
                                                    const float* __restrict__ val,
                                                    const float* __restrict__ t,
                                                    float* __restrict__ acc, int E) {
  const int idx = blockIdx.x * 256 + threadIdx.x;
  const int e = idx >> 5;
  if (e >= E) return;
  const int c = (idx & 31) * 4;
  const int r = row[e];
  const int s = col[e];
  const float v = val[e];
  const float4 tv = *reinterpret_cast<const float4*>(t + (size_t)s * HDIM + c);
  float* dst = acc + (size_t)r * HDIM + c;
  atomicAdd(dst + 0, v * tv.x);
  atomicAdd(dst + 1, v * tv.y);
  atomicAdd(dst + 2, v * tv.z);
  atomicAdd(dst + 3, v * tv.w);
}

__global__ __launch_bounds__(256) void relu_inplace(float* __restrict__ x, int n4) {
  const int i = blockIdx.x * 256 + threadIdx.x;
  if (i < n4) {
    float4* p = reinterpret_cast<float4*>(x) + i;
    float4 v = *p;
    v.x = fmaxf(v.x, 0.0f);
    v.y = fmaxf(v.y, 0.0f);
    v.z = fmaxf(v.z, 0.0f);
    v.w = fmaxf(v.w, 0.0f);
    *p = v;
  }
}

// ---------------------------------------------------------------------------
// predict = softmax(emb @ Wc, axis=1). One wave per 16 rows.
// Logits tile built with WMMA; softmax over the 16 classes reduces across the
// 16-lane N group (C/D layout: N = lane&15, groups lanes 0-15 / 16-31).
// ---------------------------------------------------------------------------
__global__ __launch_bounds__(32) void predict_softmax(const float* __restrict__ emb,
                                                      const float* __restrict__ Wc,
                                                      float* __restrict__ out) {
  const int m0 = blockIdx.x * 16;
  const int lane = threadIdx.x;
  const int m = lane & 15;
  const int n = lane & 15;
  const int kp = (lane >> 4) * 2;

  const float* arow = emb + (size_t)(m0 + m) * HDIM + kp;
  const float* wcol = Wc + n + (size_t)kp * CCLS;

  v8f acc = {};
  #pragma unroll
  for (int k = 0; k < HDIM; k += 4) {
    v2f a, b;
    a.x = arow[k];
    a.y = arow[k + 1];
    b.x = wcol[(size_t)k * CCLS];
    b.y = wcol[(size_t)(k + 1) * CCLS];
    acc = __builtin_amdgcn_wmma_f32_16x16x4_f32(false, a, false, b,
                                                (short)0, acc, false, false);
  }

  const int rowBase = m0 + (lane >> 4) * 8;
  #pragma unroll
  for (int r = 0; r < 8; ++r) {
    const float x = acc[r];
    float mx = x;
    #pragma unroll
    for (int off = 1; off < 16; off <<= 1)
      mx = fmaxf(mx, __shfl_xor(mx, off, 32));
    const float ex = expf(x - mx);
    float sum = ex;
    #pragma unroll
    for (int off = 1; off < 16; off <<= 1)
      sum += __shfl_xor(sum, off, 32);
    out[(size_t)(rowBase + r) * CCLS + n] = ex / sum;
  }
}

// ---------------------------------------------------------------------------
extern "C" void kernel_launch(void* const* d_in, const int* in_sizes, int n_in,
                              void* d_out, int out_size, void* d_ws, size_t ws_size,
                              hipStream_t stream) {
  const float* x_u    = (const float*)d_in[0];
  const float* x_v    = (const float*)d_in[1];
  const int*   row_uu = (const int*)d_in[2];
  const int*   col_uu = (const int*)d_in[3];
  const float* val_uu = (const float*)d_in[4];
  const int*   row_uv = (const int*)d_in[5];
  const int*   col_uv = (const int*)d_in[6];
  const float* val_uv = (const float*)d_in[7];
  const int*   row_vu = (const int*)d_in[8];
  const int*   col_vu = (const int*)d_in[9];
  const float* val_vu = (const float*)d_in[10];
  const float* Ws_u0  = (const float*)d_in[11];
  const float* Ws_v0  = (const float*)d_in[12];
  const float* Wm_uu0 = (const float*)d_in[13];
  const float* Wm_uv0 = (const float*)d_in[14];
  const float* Wm_vu0 = (const float*)d_in[15];
  // d_in[16..19]: attention weights -> dead (softmax over width-1 axis == 1)
  const float* Ws_u1  = (const float*)d_in[20];
  // d_in[21] Ws_v1, d_in[24] Wm_vu1: dead (layer-1 v branch unused; emb = h_u)
  const float* Wm_uu1 = (const float*)d_in[22];
  const float* Wm_uv1 = (const float*)d_in[23];
  // d_in[25..28]: layer-1 attention weights -> dead
  const float* Wc     = (const float*)d_in[29];
  const int E = in_sizes[2];

  float* emb  = (float*)d_out;                 // [NU,128]
  float* pred = emb + (size_t)NUN * HDIM;      // [NU,16]

  float* t0  = (float*)d_ws;                   // message buffers / hidden states
  float* t1  = t0  + (size_t)NUN * HDIM;
  float* t2  = t1  + (size_t)NUN * HDIM;
  float* hu1 = t2  + (size_t)NVN * HDIM;       // layer-0 pre-relu accumulators
  float* hv1 = hu1 + (size_t)NUN * HDIM;       // (relu fused into layer-1 A-stage)

  constexpr int MT = 5;                        // 80 rows/block; 100000 = 1250*80
  const dim3 gG(NUN / (16 * MT)), bG(256);
  const int scBlocks = (E * 32 + 255) / 256;

  // -------- layer 0 (A tiles staged via async global->LDS DMA) --------
  gemm_wmma<256, MT, false><<<gG, bG, 0, stream>>>(x_u, Wm_uu0, t0);
  gemm_wmma<256, MT, false><<<gG, bG, 0, stream>>>(x_v, Wm_uv0, t1);
  gemm_wmma<256, MT, false><<<gG, bG, 0, stream>>>(x_u, Wm_vu0, t2);
  gemm_wmma<256, MT, false><<<gG, bG, 0, stream>>>(x_u, Ws_u0, hu1);  // self -> acc
  gemm_wmma<256, MT, false><<<gG, bG, 0, stream>>>(x_v, Ws_v0, hv1);
  spmm_scatter<<<scBlocks, 256, 0, stream>>>(row_uu, col_uu, val_uu, t0, hu1, E);
  spmm_scatter<<<scBlocks, 256, 0, stream>>>(row_uv, col_uv, val_uv, t1, hu1, E);
  spmm_scatter<<<scBlocks, 256, 0, stream>>>(row_vu, col_vu, val_vu, t2, hv1, E);
  // hu1/hv1 are pre-relu; relu fused into the layer-1 GEMM input stage below.

  // -------- layer 1 (v branch dead: emb = h_u only) --------
  gemm_wmma<128, MT, true><<<gG, bG, 0, stream>>>(hu1, Wm_uu1, t0);
  gemm_wmma<128, MT, true><<<gG, bG, 0, stream>>>(hv1, Wm_uv1, t1);
  gemm_wmma<128, MT, true><<<gG, bG, 0, stream>>>(hu1, Ws_u1, emb);   // self -> d_out
  spmm_scatter<<<scBlocks, 256, 0, stream>>>(row_uu, col_uu, val_uu, t0, emb, E);
  spmm_scatter<<<scBlocks, 256, 0, stream>>>(row_uv, col_uv, val_uv, t1, emb, E);
  relu_inplace<<<(NUN * HDIM / 4) / 256, 256, 0, stream>>>(emb, NUN * HDIM / 4);

  // -------- classifier --------
  predict_softmax<<<NUN / 16, 32, 0, stream>>>(emb, Wc, pred);
}